// AttComms_29411936043395
// MI455X (gfx1250) — compile-verified
//
#include <hip/hip_runtime.h>
#include <hip/hip_bf16.h>
#include <math.h>

typedef float v2f __attribute__((ext_vector_type(2)));
typedef float v4f __attribute__((ext_vector_type(4)));
typedef float v8f __attribute__((ext_vector_type(8)));

#define NI 64
#define NJ 8192
#define NK 128

// -------------------------------------------------------------------------
// Kernel A: per 16-row j-tile, compute
//   ktile = x[index] @ Wk^T + bk          (WMMA f32)
//   m     = ktile @ Wq                    (WMMA f32)  -> d_ws
//   c[j]  = bq . ktile[j]                              -> d_ws
// -------------------------------------------------------------------------
__global__ __launch_bounds__(256) void proj_k_m_c(
    const float* __restrict__ x,
    const float* __restrict__ Wk, const float* __restrict__ bk,
    const float* __restrict__ Wq, const float* __restrict__ bq,
    const int*   __restrict__ index,
    float* __restrict__ m_out, float* __restrict__ c_out)
{
    __shared__ float sX [16][132];   // +4 pad: break bank conflicts on column reads
    __shared__ float sKt[16][132];

    const int j0  = blockIdx.x * 16;
    const int tid = threadIdx.x;
    const float* xi = x + (size_t)index[0] * NJ * NK;

    // stage the 16x128 tile of x[index] into LDS (coalesced)
    for (int t = tid; t < 16 * NK; t += 256)
        sX[t >> 7][t & 127] = xi[(size_t)(j0 + (t >> 7)) * NK + (t & 127)];
    __syncthreads();

    const int wave = tid >> 5;
    const int lane = tid & 31;
    const int half = lane >> 4;    // 0: K={kk,kk+1}, 1: K={kk+2,kk+3}
    const int l16  = lane & 15;    // row (A) / col (B) within 16
    const int n0   = wave * 16;    // this wave's 16-wide N slice

    // GEMM1: ktile[j,c] = sum_k Wk[c,k] * xi[j,k]   (B = Wk^T -> B[k][n]=Wk[n][k])
    v8f acc = {};
    for (int kk = 0; kk < NK; kk += 4) {
        const int ka = kk + 2 * half;
        v2f a, b;
        a.x = sX[l16][ka + 0];
        a.y = sX[l16][ka + 1];
        b.x = Wk[(n0 + l16) * NK + ka + 0];
        b.y = Wk[(n0 + l16) * NK + ka + 1];
        acc = __builtin_amdgcn_wmma_f32_16x16x4_f32(false, a, false, b,
                                                    (short)0, acc, false, false);
    }
    const float bkn = bk[n0 + l16];
#pragma unroll
    for (int r = 0; r < 8; ++r)            // D: row = r + 8*half, col = n0+l16
        sKt[r + 8 * half][n0 + l16] = acc[r] + bkn;
    __syncthreads();

    // GEMM2: m = ktile @ Wq   (B[k][n] = Wq[k][n])
    v8f acc2 = {};
    for (int kk = 0; kk < NK; kk += 4) {
        const int ka = kk + 2 * half;
        v2f a, b;
        a.x = sKt[l16][ka + 0];
        a.y = sKt[l16][ka + 1];
        b.x = Wq[(ka + 0) * NK + n0 + l16];
        b.y = Wq[(ka + 1) * NK + n0 + l16];
        acc2 = __builtin_amdgcn_wmma_f32_16x16x4_f32(false, a, false, b,
                                                     (short)0, acc2, false, false);
    }
#pragma unroll
    for (int r = 0; r < 8; ++r)
        m_out[(size_t)(j0 + r + 8 * half) * NK + n0 + l16] = acc2[r];

    // c[j] = bq . ktile[j] : wave w reduces rows 2w (lanes 0-15) and 2w+1 (16-31)
    const int jl = 2 * wave + half;
    float p = 0.f;
#pragma unroll
    for (int c = 0; c < NK; c += 16)
        p += bq[c + l16] * sKt[jl][c + l16];
    p += __shfl_xor(p, 8);
    p += __shfl_xor(p, 4);
    p += __shfl_xor(p, 2);
    p += __shfl_xor(p, 1);
    if (l16 == 0) c_out[j0 + jl] = p;
}

// -------------------------------------------------------------------------
// Kernel B: single streaming pass over x (256 MB, read once, non-temporal).
// One wave per j. Online softmax over i with fused weighted accumulation:
//   s[j] = sum_i softmax_i( x[i,j].m[j] + c[j] ) * x[i,j]
// -------------------------------------------------------------------------
__global__ __launch_bounds__(256) void fused_softmax_s(
    const float* __restrict__ x,
    const float* __restrict__ m,
    const float* __restrict__ cvec,
    float* __restrict__ s_out)
{
    const int wave = threadIdx.x >> 5;
    const int lane = threadIdx.x & 31;
    const int j    = blockIdx.x * 8 + wave;

    const v4f* mrow = (const v4f*)(m + (size_t)j * NK);
    const v4f  mv   = mrow[lane];
    const float cj  = cvec[j];

    v4f   sacc    = {};
    float run_max = -INFINITY;
    float run_sum = 0.f;

    const v4f* xbase = (const v4f*)x;
    const size_t row = (size_t)j * (NK / 4) + lane;

    for (int i = 0; i < NI; ++i) {
        v4f xv = __builtin_nontemporal_load(xbase + (size_t)i * NJ * (NK / 4) + row);
        float p = xv.x * mv.x + xv.y * mv.y + xv.z * mv.z + xv.w * mv.w;
        p += __shfl_xor(p, 16);
        p += __shfl_xor(p, 8);
        p += __shfl_xor(p, 4);
        p += __shfl_xor(p, 2);
        p += __shfl_xor(p, 1);
        const float logit = p + cj;
        const float nm = fmaxf(run_max, logit);
        const float sc = __expf(run_max - nm);   // rescale old accumulators
        const float w  = __expf(logit  - nm);    // weight of this agent
        run_max = nm;
        run_sum = run_sum * sc + w;
        sacc    = sacc * sc + xv * w;
    }
    const float inv = 1.0f / run_sum;
    ((v4f*)(s_out + (size_t)j * NK))[lane] = sacc * inv;
}

// -------------------------------------------------------------------------
// Kernel C: features = s @ Wv^T + bv   (WMMA f32) -> d_out
// (exact because sum_i att[i,j] == 1 after softmax)
// -------------------------------------------------------------------------
__global__ __launch_bounds__(256) void out_gemm(
    const float* __restrict__ s,
    const float* __restrict__ Wv, const float* __restrict__ bv,
    float* __restrict__ out)
{
    __shared__ float sS[16][132];
    const int j0  = blockIdx.x * 16;
    const int tid = threadIdx.x;
    for (int t = tid; t < 16 * NK; t += 256)
        sS[t >> 7][t & 127] = s[(size_t)(j0 + (t >> 7)) * NK + (t & 127)];
    __syncthreads();

    const int wave = tid >> 5;
    const int lane = tid & 31;
    const int half = lane >> 4;
    const int l16  = lane & 15;
    const int n0   = wave * 16;

    v8f acc = {};
    for (int kk = 0; kk < NK; kk += 4) {
        const int ka = kk + 2 * half;
        v2f a, b;
        a.x = sS[l16][ka + 0];
        a.y = sS[l16][ka + 1];
        b.x = Wv[(n0 + l16) * NK + ka + 0];   // B = Wv^T
        b.y = Wv[(n0 + l16) * NK + ka + 1];
        acc = __builtin_amdgcn_wmma_f32_16x16x4_f32(false, a, false, b,
                                                    (short)0, acc, false, false);
    }
    const float bvn = bv[n0 + l16];
#pragma unroll
    for (int r = 0; r < 8; ++r)
        out[(size_t)(j0 + r + 8 * half) * NK + n0 + l16] = acc[r] + bvn;
}

// -------------------------------------------------------------------------
extern "C" void kernel_launch(void* const* d_in, const int* in_sizes, int n_in,
                              void* d_out, int out_size, void* d_ws, size_t ws_size,
                              hipStream_t stream)
{
    const float* x     = (const float*)d_in[0];
    const float* Wv    = (const float*)d_in[1];
    const float* bv    = (const float*)d_in[2];
    const float* Wq    = (const float*)d_in[3];
    const float* bq    = (const float*)d_in[4];
    const float* Wk    = (const float*)d_in[5];
    const float* bk    = (const float*)d_in[6];
    const int*   index = (const int*)d_in[7];

    float* m    = (float*)d_ws;                 // NJ*NK floats (4 MB)
    float* cvec = m + (size_t)NJ * NK;          // NJ floats
    float* sbuf = cvec + NJ;                    // NJ*NK floats (4 MB)
    float* out  = (float*)d_out;

    proj_k_m_c     <<<NJ / 16, 256, 0, stream>>>(x, Wk, bk, Wq, bq, index, m, cvec);
    fused_softmax_s<<<NJ / 8,  256, 0, stream>>>(x, m, cvec, sbuf);
    out_gemm       <<<NJ / 16, 256, 0, stream>>>(sbuf, Wv, bv, out);
}